// GAT_55972013801741
// MI455X (gfx1250) — compile-verified
//
#include <hip/hip_runtime.h>
#include <math.h>

// ---------------------------------------------------------------------------
// 2-layer GAT + linear head for MI455X (gfx1250, wave32, WGP, 320KB LDS).
// Dense GEMMs: V_WMMA_F32_16X16X4_F32 fed from LDS-staged panels (padded to
// kill bank conflicts), staged with GLOBAL_LOAD_ASYNC_TO_LDS_B128 (ASYNCcnt)
// when the toolchain exposes the builtin.  Edge phase: f32 global atomics;
// the 102MB feature tensors are L2-resident (192MB L2) so random edge
// traffic stays on-chip.
// ---------------------------------------------------------------------------

typedef __attribute__((ext_vector_type(2))) float v2f;
typedef __attribute__((ext_vector_type(8))) float v8f;
typedef __attribute__((ext_vector_type(4))) int   v4i;

#define NEG_SLOPE 0.2f
#define ORD_NEG_INF 0x007FFFFFu   // f2ord(-inf)

#if __has_builtin(__builtin_amdgcn_global_load_async_to_lds_b128)
#define HAVE_ASYNC_LDS 1
#endif

__device__ __forceinline__ void stage_b128(const float4* g, float4* l) {
#ifdef HAVE_ASYNC_LDS
  // (global v4i* src, local v4i* dst, imm offset, imm cpol)
  __builtin_amdgcn_global_load_async_to_lds_b128(
      (__attribute__((address_space(1))) v4i*)g,
      (__attribute__((address_space(3))) v4i*)l, 0, 0);
#else
  *l = *g;
#endif
}

__device__ __forceinline__ void stage_wait() {
#ifdef HAVE_ASYNC_LDS
#if __has_builtin(__builtin_amdgcn_s_wait_asynccnt)
  __builtin_amdgcn_s_wait_asynccnt(0);
#else
  asm volatile("s_wait_asynccnt 0" ::: "memory");
#endif
#endif
}

__device__ __forceinline__ unsigned f2ord(float f) {
  unsigned u = __float_as_uint(f);
  return (u & 0x80000000u) ? ~u : (u | 0x80000000u);
}
__device__ __forceinline__ float ord2f(unsigned v) {
  return (v & 0x80000000u) ? __uint_as_float(v & 0x7FFFFFFFu)
                           : __uint_as_float(~v);
}

// ---------------------------------------------------------------------------
// GEMM1: C[M,512] = A[M,256] @ B[256,512].  One block per 16-row M-tile,
// 8 waves x 64 output cols.  A staged fully in LDS (stride 260), B staged in
// 32-deep k-chunks (stride 516).  One A fragment feeds 4 WMMAs.
// ---------------------------------------------------------------------------
#define LA_STRIDE 260   // 256 + 4 pad (floats)
#define LB_STRIDE 516   // 512 + 4 pad (floats)

__global__ void __launch_bounds__(256) wmma_gemm1(const float* __restrict__ A,
                                                  const float* __restrict__ B,
                                                  float* __restrict__ C, int M) {
  __shared__ float lA[16 * LA_STRIDE];   // 16.6 KB
  __shared__ float lB[32 * LB_STRIDE];   // 66.0 KB

  const int t    = threadIdx.x;
  const int w    = t >> 5;        // wave 0..7
  const int lane = t & 31;
  const int half = lane >> 4;
  const int l16  = lane & 15;
  const int m0   = blockIdx.x * 16;

  // ---- stage A tile (16 x 256) as float4, row-clamped ----
  {
    const float4* Av4 = (const float4*)A;
    float4* lAv4 = (float4*)lA;
#pragma unroll
    for (int i = 0; i < 4; ++i) {
      const int idx4 = t + i * 256;           // 0..1023
      const int row  = idx4 >> 6;             // /64 float4 per row
      const int col4 = idx4 & 63;
      const int grow = m0 + row < M ? m0 + row : M - 1;
      stage_b128(&Av4[(size_t)grow * 64 + col4], &lAv4[row * (LA_STRIDE / 4) + col4]);
    }
  }

  v8f acc[4];
#pragma unroll
  for (int j = 0; j < 4; ++j) acc[j] = (v8f){0.f,0.f,0.f,0.f,0.f,0.f,0.f,0.f};

  const float4* Bv4 = (const float4*)B;
  float4* lBv4 = (float4*)lB;

  for (int k0 = 0; k0 < 256; k0 += 32) {
    if (k0) __syncthreads();   // previous chunk fully consumed
    // ---- stage B chunk (32 x 512) as float4, coalesced ----
#pragma unroll
    for (int i = 0; i < 16; ++i) {
      const int idx4 = t + i * 256;           // 0..4095
      const int row  = idx4 >> 7;             // /128 float4 per row
      const int col4 = idx4 & 127;
      stage_b128(&Bv4[(size_t)(k0 + row) * 128 + col4],
                 &lBv4[row * (LB_STRIDE / 4) + col4]);
    }
    if (k0 + 32 < 256)  // prefetch next chunk into L2 (global_prefetch_b8)
      __builtin_prefetch(&B[(size_t)(k0 + 32) * 512 + t * 8], 0, 1);
    stage_wait();
    __syncthreads();

#pragma unroll
    for (int kk = 0; kk < 32; kk += 4) {
      const int ka = kk + 2 * half;
      v2f a;
      a.x = lA[l16 * LA_STRIDE + k0 + ka + 0];
      a.y = lA[l16 * LA_STRIDE + k0 + ka + 1];
#pragma unroll
      for (int j = 0; j < 4; ++j) {
        const int cb = w * 64 + j * 16 + l16;
        v2f b;
        b.x = lB[(ka + 0) * LB_STRIDE + cb];
        b.y = lB[(ka + 1) * LB_STRIDE + cb];
        acc[j] = __builtin_amdgcn_wmma_f32_16x16x4_f32(false, a, false, b,
                                                       (short)0, acc[j], false, false);
      }
    }
  }

#pragma unroll
  for (int j = 0; j < 4; ++j) {
    const int col = w * 64 + j * 16 + l16;
#pragma unroll
    for (int r = 0; r < 8; ++r) {
      const int orow = m0 + r + 8 * half;
      if (orow < M) C[(size_t)orow * 512 + col] = acc[j][r];
    }
  }
}

// ---------------------------------------------------------------------------
// GEMM2: C[M,32] = A[M,512] @ B[512,32].  Full B staged in LDS (stride 36);
// block = 8 waves, each wave one 16-row M-tile x 32 cols (2 WMMA tiles).
// ---------------------------------------------------------------------------
#define LB2_STRIDE 36   // 32 + 4 pad

__global__ void __launch_bounds__(256) wmma_gemm2(const float* __restrict__ A,
                                                  const float* __restrict__ B,
                                                  float* __restrict__ C, int M) {
  __shared__ float lB[512 * LB2_STRIDE];   // 73.7 KB

  const int t    = threadIdx.x;
  const int w    = t >> 5;
  const int lane = t & 31;
  const int half = lane >> 4;
  const int l16  = lane & 15;

  // ---- stage B (512 x 32) as float4 ----
  {
    const float4* Bv4 = (const float4*)B;
    float4* lBv4 = (float4*)lB;
#pragma unroll
    for (int i = 0; i < 16; ++i) {
      const int idx4 = t + i * 256;           // 0..4095
      const int row  = idx4 >> 3;             // /8 float4 per row
      const int col4 = idx4 & 7;
      stage_b128(&Bv4[(size_t)row * 8 + col4], &lBv4[row * (LB2_STRIDE / 4) + col4]);
    }
  }
  stage_wait();
  __syncthreads();

  const int m0 = blockIdx.x * 128 + w * 16;
  if (m0 < M) {   // wave-uniform guard; no barriers below
    const int row  = m0 + l16;
    const int rowc = row < M ? row : M - 1;
    const float* __restrict__ arow = A + (size_t)rowc * 512;

    v8f acc0 = {0.f,0.f,0.f,0.f,0.f,0.f,0.f,0.f};
    v8f acc1 = acc0;
    for (int k0 = 0; k0 < 512; k0 += 4) {
      const int ka = k0 + 2 * half;
      v2f a, b0, b1;
      a.x  = arow[ka + 0];
      a.y  = arow[ka + 1];
      b0.x = lB[(ka + 0) * LB2_STRIDE + l16];
      b0.y = lB[(ka + 1) * LB2_STRIDE + l16];
      b1.x = lB[(ka + 0) * LB2_STRIDE + 16 + l16];
      b1.y = lB[(ka + 1) * LB2_STRIDE + 16 + l16];
      acc0 = __builtin_amdgcn_wmma_f32_16x16x4_f32(false, a, false, b0,
                                                   (short)0, acc0, false, false);
      acc1 = __builtin_amdgcn_wmma_f32_16x16x4_f32(false, a, false, b1,
                                                   (short)0, acc1, false, false);
    }
#pragma unroll
    for (int r = 0; r < 8; ++r) {
      const int orow = m0 + r + 8 * half;
      if (orow < M) {
        C[(size_t)orow * 32 + l16]      = acc0[r];
        C[(size_t)orow * 32 + 16 + l16] = acc1[r];
      }
    }
  }
}

// ---------------------------------------------------------------------------
// FC head: out[M,40] = H[M,32] @ W[32,40] + bias.  3 padded col tiles (48);
// clamped loads keep EXEC full for WMMA, stores guarded to col<40.
// ---------------------------------------------------------------------------
__global__ void wmma_fc_f32(const float* __restrict__ H,
                            const float* __restrict__ Wm,
                            const float* __restrict__ bias,
                            float* __restrict__ out, int M) {
  const int wave = blockIdx.x * (blockDim.x >> 5) + (threadIdx.x >> 5);
  const int lane = threadIdx.x & 31;
  const int mt   = wave / 3;
  const int nt   = wave - mt * 3;
  if (mt * 16 >= M) return;

  const int half = lane >> 4;
  const int l16  = lane & 15;
  const int row  = mt * 16 + l16;
  const int rowc = row < M ? row : M - 1;
  const int col  = nt * 16 + l16;
  const int colc = col < 40 ? col : 39;

  const float* __restrict__ hrow = H + (size_t)rowc * 32;
  v8f acc = {0.f,0.f,0.f,0.f,0.f,0.f,0.f,0.f};

#pragma unroll
  for (int k0 = 0; k0 < 32; k0 += 4) {
    const int ka = k0 + 2 * half;
    v2f a, b;
    a.x = hrow[ka + 0];
    a.y = hrow[ka + 1];
    b.x = Wm[(ka + 0) * 40 + colc];
    b.y = Wm[(ka + 1) * 40 + colc];
    acc = __builtin_amdgcn_wmma_f32_16x16x4_f32(false, a, false, b,
                                                (short)0, acc, false, false);
  }
  if (col < 40) {
    const float bb = bias[col];
#pragma unroll
    for (int r = 0; r < 8; ++r) {
      const int orow = mt * 16 + r + 8 * half;
      if (orow < M) out[(size_t)orow * 40 + col] = acc[r] + bb;
    }
  }
}

// ---------------------------------------------------------------------------
// Attention coefficient dots.
// ---------------------------------------------------------------------------
__global__ void att_dots_l1(const float* __restrict__ xl,
                            const float* __restrict__ att_s,
                            const float* __restrict__ att_d,
                            float* __restrict__ a_s, float* __restrict__ a_d,
                            int n) {
  const int idx = blockIdx.x * blockDim.x + threadIdx.x;  // (node, head)
  if (idx >= n * 8) return;
  const int node = idx >> 3, h = idx & 7;
  const float* __restrict__ v = xl + (size_t)node * 512 + h * 64;
  const float* __restrict__ s = att_s + h * 64;
  const float* __restrict__ d = att_d + h * 64;
  float ss = 0.f, dd = 0.f;
#pragma unroll 8
  for (int c = 0; c < 64; ++c) {
    const float x = v[c];
    ss += x * s[c];
    dd += x * d[c];
  }
  a_s[idx] = ss;
  a_d[idx] = dd;
}

__global__ void att_dots_l2(const float* __restrict__ xl,
                            const float* __restrict__ att_s,
                            const float* __restrict__ att_d,
                            float* __restrict__ a_s, float* __restrict__ a_d,
                            int n) {
  const int idx = blockIdx.x * blockDim.x + threadIdx.x;
  if (idx >= n) return;
  const float* __restrict__ v = xl + (size_t)idx * 32;
  float ss = 0.f, dd = 0.f;
#pragma unroll 8
  for (int c = 0; c < 32; ++c) {
    const float x = v[c];
    ss += x * att_s[c];
    dd += x * att_d[c];
  }
  a_s[idx] = ss;
  a_d[idx] = dd;
}

// ---------------------------------------------------------------------------
// Segment softmax over incoming edges (self-loops appended logically at e>=E).
// ---------------------------------------------------------------------------
__global__ void init_neginf(unsigned* __restrict__ a, int na,
                            unsigned* __restrict__ b, int nb) {
  const int i = blockIdx.x * blockDim.x + threadIdx.x;
  if (i < na) a[i] = ORD_NEG_INF;
  else if (i - na < nb) b[i - na] = ORD_NEG_INF;
}

__global__ void edge_leaky_max(const int* __restrict__ ei, int E, int n, int H,
                               const float* __restrict__ a_s,
                               const float* __restrict__ a_d,
                               unsigned* __restrict__ emax) {
  const int idx = blockIdx.x * blockDim.x + threadIdx.x;
  if (idx >= (E + n) * H) return;
  const int e = idx / H, h = idx - e * H;
  int s, d;
  if (e < E) { s = ei[e]; d = ei[E + e]; } else { s = e - E; d = s; }
  float v = a_s[s * H + h] + a_d[d * H + h];
  v = v > 0.f ? v : NEG_SLOPE * v;
  atomicMax(&emax[d * H + h], f2ord(v));
}

__global__ void edge_exp_sum(const int* __restrict__ ei, int E, int n, int H,
                             const float* __restrict__ a_s,
                             const float* __restrict__ a_d,
                             const unsigned* __restrict__ emax,
                             float* __restrict__ p, float* __restrict__ denom) {
  const int idx = blockIdx.x * blockDim.x + threadIdx.x;
  if (idx >= (E + n) * H) return;
  const int e = idx / H, h = idx - e * H;
  int s, d;
  if (e < E) { s = ei[e]; d = ei[E + e]; } else { s = e - E; d = s; }
  float v = a_s[s * H + h] + a_d[d * H + h];
  v = v > 0.f ? v : NEG_SLOPE * v;
  float m = ord2f(emax[d * H + h]);
  if (!isfinite(m)) m = 0.f;                 // reference: where(isfinite, emax, 0)
  const float pe = expf(v - m);
  p[idx] = pe;
  atomicAdd(&denom[d * H + h], pe);
}

// Layer-1 scatter: one block per edge, 512 threads = (head, channel).
__global__ void scatter_l1(const int* __restrict__ ei, int E, int n,
                           const float* __restrict__ xl,
                           const float* __restrict__ p,
                           const float* __restrict__ denom,
                           float* __restrict__ out) {
  const int e = blockIdx.x;
  const int t = threadIdx.x;          // h*64 + c
  const int h = t >> 6;
  int s, d;
  if (e < E) { s = ei[e]; d = ei[E + e]; } else { s = e - E; d = s; }
  const float alpha = p[(size_t)e * 8 + h] / (denom[d * 8 + h] + 1e-16f);
  atomicAdd(&out[(size_t)d * 512 + t], xl[(size_t)s * 512 + t] * alpha);
}

// Layer-2 scatter: thread per (edge, channel), H == 1, C == 32.
__global__ void scatter_l2(const int* __restrict__ ei, int E, int n,
                           const float* __restrict__ xl,
                           const float* __restrict__ p,
                           const float* __restrict__ denom,
                           float* __restrict__ out) {
  const int idx = blockIdx.x * blockDim.x + threadIdx.x;
  if (idx >= (E + n) * 32) return;
  const int e = idx >> 5, c = idx & 31;
  int s, d;
  if (e < E) { s = ei[e]; d = ei[E + e]; } else { s = e - E; d = s; }
  const float alpha = p[e] / (denom[d] + 1e-16f);
  atomicAdd(&out[(size_t)d * 32 + c], xl[(size_t)s * 32 + c] * alpha);
}

__global__ void bias_relu(const float* __restrict__ in,
                          const float* __restrict__ bias,
                          float* __restrict__ out, int n, int C) {
  const int idx = blockIdx.x * blockDim.x + threadIdx.x;
  if (idx >= n * C) return;
  const float v = in[idx] + bias[idx % C];
  out[idx] = v > 0.f ? v : 0.f;
}

// ---------------------------------------------------------------------------
// Launcher
// ---------------------------------------------------------------------------
extern "C" void kernel_launch(void* const* d_in, const int* in_sizes, int n_in,
                              void* d_out, int out_size, void* d_ws,
                              size_t ws_size, hipStream_t stream) {
  const float* x   = (const float*)d_in[0];
  const int*   ei  = (const int*)d_in[1];
  const float* W1  = (const float*)d_in[2];
  const float* as1 = (const float*)d_in[3];
  const float* ad1 = (const float*)d_in[4];
  const float* b1  = (const float*)d_in[5];
  const float* W2  = (const float*)d_in[6];
  const float* as2 = (const float*)d_in[7];
  const float* ad2 = (const float*)d_in[8];
  const float* b2  = (const float*)d_in[9];
  const float* fcW = (const float*)d_in[10];
  const float* fcb = (const float*)d_in[11];

  const int N  = in_sizes[0] / 256;   // 50000
  const int E  = in_sizes[1] / 2;     // 800000
  const int ET = E + N;               // with self loops

  // ---- workspace layout (floats) ----
  float* ws = (float*)d_ws;
  size_t off = 0;
  float*    xl1    = ws + off; off += (size_t)N * 512;
  float*    out1   = ws + off; off += (size_t)N * 512;   // becomes h1 in-place
  float*    a_src1 = ws + off; off += (size_t)N * 8;
  float*    a_dst1 = ws + off; off += (size_t)N * 8;
  unsigned* emax1  = (unsigned*)(ws + off); off += (size_t)N * 8;
  float*    den1   = ws + off; off += (size_t)N * 8;
  float*    p1     = ws + off; off += (size_t)ET * 8;
  float*    xl2    = ws + off; off += (size_t)N * 32;
  float*    out2   = ws + off; off += (size_t)N * 32;
  float*    a_src2 = ws + off; off += (size_t)N;
  float*    a_dst2 = ws + off; off += (size_t)N;
  unsigned* emax2  = (unsigned*)(ws + off); off += (size_t)N;
  float*    den2   = ws + off; off += (size_t)N;
  float*    p2     = ws + off; off += (size_t)ET;

  float* emb    = (float*)d_out;                   // [N,32]
  float* logits = (float*)d_out + (size_t)N * 32;  // [N,40]

  const int B = 256;
  // ---- per-launch re-init (graph replay safe) ----
  (void)hipMemsetAsync(out1, 0, (size_t)N * 512 * sizeof(float), stream);
  (void)hipMemsetAsync(den1, 0, (size_t)N * 8 * sizeof(float), stream);
  (void)hipMemsetAsync(out2, 0, (size_t)N * 32 * sizeof(float), stream);
  (void)hipMemsetAsync(den2, 0, (size_t)N * sizeof(float), stream);
  {
    const int tot = N * 9;
    init_neginf<<<(tot + B - 1) / B, B, 0, stream>>>(emax1, N * 8, emax2, N);
  }

  const int mtiles = (N + 15) / 16;

  // ---- layer 1 ----
  wmma_gemm1<<<mtiles, 256, 0, stream>>>(x, W1, xl1, N);
  att_dots_l1<<<(N * 8 + B - 1) / B, B, 0, stream>>>(xl1, as1, ad1, a_src1, a_dst1, N);
  edge_leaky_max<<<(ET * 8 + B - 1) / B, B, 0, stream>>>(ei, E, N, 8, a_src1, a_dst1, emax1);
  edge_exp_sum<<<(ET * 8 + B - 1) / B, B, 0, stream>>>(ei, E, N, 8, a_src1, a_dst1, emax1, p1, den1);
  scatter_l1<<<ET, 512, 0, stream>>>(ei, E, N, xl1, p1, den1, out1);
  bias_relu<<<(N * 512 + B - 1) / B, B, 0, stream>>>(out1, b1, out1, N, 512); // -> h1

  // ---- layer 2 ----
  wmma_gemm2<<<(mtiles + 7) / 8, 256, 0, stream>>>(out1, W2, xl2, N);
  att_dots_l2<<<(N + B - 1) / B, B, 0, stream>>>(xl2, as2, ad2, a_src2, a_dst2, N);
  edge_leaky_max<<<(ET + B - 1) / B, B, 0, stream>>>(ei, E, N, 1, a_src2, a_dst2, emax2);
  edge_exp_sum<<<(ET + B - 1) / B, B, 0, stream>>>(ei, E, N, 1, a_src2, a_dst2, emax2, p2, den2);
  scatter_l2<<<(ET * 32 + B - 1) / B, B, 0, stream>>>(ei, E, N, xl2, p2, den2, out2);
  bias_relu<<<(N * 32 + B - 1) / B, B, 0, stream>>>(out2, b2, emb, N, 32); // embedding

  // ---- FC head ----
  {
    const int waves = mtiles * 3;  // 40 cols -> 3 padded tiles
    wmma_fc_f32<<<(waves * 32 + B - 1) / B, B, 0, stream>>>(emb, fcW, fcb, logits, N);
  }
}